// Efficient8BitALU_AddSub_7945689497929
// MI455X (gfx1250) — compile-verified
//
#include <hip/hip_runtime.h>

typedef __attribute__((ext_vector_type(2))) float v2f;
typedef __attribute__((ext_vector_type(8))) float v8f;

// ---- problem constants (match reference) ----
#define D_BD      128
#define ROWP      132     // padded LDS row stride (floats): breaks 64-bank aliasing
#define TOKS      16      // tokens per wave-tile (one WMMA M dimension)
#define WAVES_PB  4       // waves per block (128 threads)
#define HID_TILES 8       // 128 hidden / 16

// bd layout offsets
#define OFF_MARK   0
#define OFF_ADD    1
#define OFF_SUB    2
#define OFF_ALULO  3
#define OFF_ALUHI  19
#define OFF_CARLO  35
#define OFF_CARHI  51
#define OFF_OUTLO  67
#define OFF_OUTHI  83
#define OFF_OP25   124    // OPCODE_BASE(99)+25
#define OFF_OP26   125    // OPCODE_BASE(99)+26

#if __has_builtin(__builtin_amdgcn_update_dpp)
#define HAVE_DPP16 1
#endif

#if __has_builtin(__builtin_amdgcn_sched_barrier)
#define SCHED_FENCE() __builtin_amdgcn_sched_barrier(0)
#else
#define SCHED_FENCE() do {} while (0)
#endif

// Interleave pattern for one hidden tile: (1 WMMA, ~20 consumer VALUs) x 4 groups.
// Keeps only ~1 D-tile + one group's relu temps live at a time.
#if __has_builtin(__builtin_amdgcn_sched_group_barrier)
#define TILE_CHOREO()                                            \
    do {                                                         \
        _Pragma("unroll")                                        \
        for (int _g = 0; _g < 4; ++_g) {                         \
            __builtin_amdgcn_sched_group_barrier(0x008, 1, 0);   \
            __builtin_amdgcn_sched_group_barrier(0x002, 20, 0);  \
        }                                                        \
    } while (0)
#else
#define TILE_CHOREO() do {} while (0)
#endif

#ifdef HAVE_DPP16
// x += (x shifted right by SH lanes within each 16-lane row, zeros shifted in)
template <int CTRL>
__device__ __forceinline__ float dpp_shr_add(float x) {
    const int y = __builtin_amdgcn_update_dpp(0, __float_as_int(x), CTRL,
                                              0xF, 0xF, /*bound_ctrl=*/true);
    return x + __int_as_float(y);
}
#endif

// After this, lane 15 of each 16-lane row holds the sum of its row's 16 values.
__device__ __forceinline__ float rowsum16(float x) {
#ifdef HAVE_DPP16
    x = dpp_shr_add<0x111>(x);   // row_shr:1
    x = dpp_shr_add<0x112>(x);   // row_shr:2
    x = dpp_shr_add<0x114>(x);   // row_shr:4
    x = dpp_shr_add<0x118>(x);   // row_shr:8  -> lane15 = row total
    return x;
#else
    #pragma unroll
    for (int off = 1; off < 16; off <<= 1) x += __shfl_xor(x, off, 32);
    return x;                     // all lanes hold the total
#endif
}

__global__ __launch_bounds__(WAVES_PB * 32) void alu_wmma_kernel(
    const float* __restrict__ x,
    const float* __restrict__ Wa1, const float* __restrict__ ba1g,
    const float* __restrict__ Wa2, const float* __restrict__ ba2g,
    const float* __restrict__ Ws1, const float* __restrict__ bs1g,
    const float* __restrict__ Ws2, const float* __restrict__ bs2g,
    float* __restrict__ out, int numTiles)
{
    __shared__ float lds[WAVES_PB * TOKS * ROWP];   // per-wave token staging
    __shared__ float red[WAVES_PB][4][TOKS];        // per-wave reduced MLP results

    const int lane = threadIdx.x & 31;
    const int wave = threadIdx.x >> 5;
    const int gwave = blockIdx.x * WAVES_PB + wave;
    const int nwaves = gridDim.x * WAVES_PB;
    float* myLds = lds + wave * TOKS * ROWP;

    const int half = lane >> 4;       // 0: lanes 0-15 (K0/K1), 1: lanes 16-31 (K2/K3)
    const int l15  = lane & 15;

    // ---- preload B-matrices (4 live rows of W1), W2 column 63, biases ----
    // B 4x16 f32 layout (mirror of A 16x4): VGPR0 = K0 (lanes0-15) / K2 (lanes16-31),
    //                                       VGPR1 = K1 (lanes0-15) / K3 (lanes16-31)
    const int kFirst  = half ? 27 : 0;    // K0 -> W1 row 0 (NIB_A),  K2 -> row 27 (op25)
    const int kSecond = half ? 28 : 1;    // K1 -> W1 row 1 (NIB_B),  K3 -> row 28 (op26)
    v2f Badd[HID_TILES], Bsub[HID_TILES];
    float w2a[HID_TILES], w2s[HID_TILES], b1a[HID_TILES], b1s[HID_TILES];
    #pragma unroll
    for (int t = 0; t < HID_TILES; ++t) {
        const int n = t * 16 + l15;                  // hidden index owned by this lane
        Badd[t].x = Wa1[kFirst * 128 + n];
        Badd[t].y = Wa1[kSecond * 128 + n];
        Bsub[t].x = Ws1[kFirst * 128 + n];
        Bsub[t].y = Ws1[kSecond * 128 + n];
        w2a[t] = Wa2[n * 64 + 63];                   // W2[:,RESULT]
        w2s[t] = Ws2[n * 64 + 63];
        b1a[t] = ba1g[n];
        b1s[t] = bs1g[n];
    }
    const float b2a = ba2g[63];
    const float b2s = bs2g[63];

    for (int tile = gwave; tile < numTiles; tile += nwaves) {
        const float* src = x   + (size_t)tile * TOKS * D_BD;
        float*       dst = out + (size_t)tile * TOKS * D_BD;

        // ---- stage 16 tokens x 128 floats into padded LDS rows (coalesced b128) ----
        #pragma unroll
        for (int i = 0; i < TOKS; ++i) {
            const float4 v = *(const float4*)(src + i * D_BD + lane * 4);
            *(float4*)(myLds + i * ROWP + lane * 4) = v;
        }

        // ---- nibble decode: first k in 0..15 with v>0.5 else 0 ----
        const int tok = l15;
        const float* row = myLds + tok * ROWP;
        const int baseA = half ? OFF_ALUHI : OFF_ALULO;
        const int baseB = half ? OFF_CARHI : OFF_CARLO;
        int ia = 0, ib = 0;
        #pragma unroll
        for (int k = 15; k >= 0; --k) {              // descending: smallest hit wins
            if (row[baseA + k] > 0.5f) ia = k;
            if (row[baseB + k] > 0.5f) ib = k;
        }
        const float dA = (float)ia, dB = (float)ib;
        const float oA = __shfl_xor(dA, 16, 32);     // lanes0-15 receive hi-decodes
        const float oB = __shfl_xor(dB, 16, 32);
        const float v25 = row[OFF_OP25];
        const float v26 = row[OFF_OP26];

        // A matrices (16x4 f32): lanes0-15 {K0,K1} = nibbles, lanes16-31 {K2,K3} = ops
        v2f Alo, Ahi;
        Alo.x = half ? v25 : dA;  Alo.y = half ? v26 : dB;
        Ahi.x = half ? v25 : oA;  Ahi.y = half ? v26 : oB;

        // ---- layer1 via WMMA (K=4 == full contraction), fused bias+ReLU+dot(W2col) ----
        v8f accLA = {}, accLS = {}, accHA = {}, accHS = {};
        #pragma unroll
        for (int t = 0; t < HID_TILES; ++t) {
            const v8f c = {};
            v8f d;
            d = __builtin_amdgcn_wmma_f32_16x16x4_f32(false, Alo, false, Badd[t], (short)0, c, false, false);
            #pragma unroll
            for (int v = 0; v < 8; ++v) accLA[v] += fmaxf(d[v] + b1a[t], 0.f) * w2a[t];
            d = __builtin_amdgcn_wmma_f32_16x16x4_f32(false, Alo, false, Bsub[t], (short)0, c, false, false);
            #pragma unroll
            for (int v = 0; v < 8; ++v) accLS[v] += fmaxf(d[v] + b1s[t], 0.f) * w2s[t];
            d = __builtin_amdgcn_wmma_f32_16x16x4_f32(false, Ahi, false, Badd[t], (short)0, c, false, false);
            #pragma unroll
            for (int v = 0; v < 8; ++v) accHA[v] += fmaxf(d[v] + b1a[t], 0.f) * w2a[t];
            d = __builtin_amdgcn_wmma_f32_16x16x4_f32(false, Ahi, false, Bsub[t], (short)0, c, false, false);
            #pragma unroll
            for (int v = 0; v < 8; ++v) accHS[v] += fmaxf(d[v] + b1s[t], 0.f) * w2s[t];
            TILE_CHOREO();   // (1 WMMA, 20 VALU) x 4: consumers retire before next D-tile
            SCHED_FENCE();   // and nothing migrates across hidden-tile boundaries
        }

        // ---- VALU-only 16-lane reduction (DPP row_shr scan; no LDS traffic) ----
        #pragma unroll
        for (int v = 0; v < 8; ++v) {
            accLA[v] = rowsum16(accLA[v]);
            accLS[v] = rowsum16(accLS[v]);
            accHA[v] = rowsum16(accHA[v]);
            accHS[v] = rowsum16(accHS[v]);
        }
        // D-tile M mapping: VGPR v <-> token v (lanes0-15) / token 8+v (lanes16-31).
        // After rowsum16, lane 15 of each row holds the row total.
        if (l15 == 15) {
            #pragma unroll
            for (int v = 0; v < 8; ++v) {
                red[wave][0][half * 8 + v] = accLA[v];   // lo,add
                red[wave][1][half * 8 + v] = accLS[v];   // lo,sub
                red[wave][2][half * 8 + v] = accHA[v];   // hi,add
                red[wave][3][half * 8 + v] = accHS[v];   // hi,sub
            }
        }

        // ---- finalize: lane handles (token=l15, position=half) ----
        {
            const float active = row[OFF_MARK];
            const float isadd  = row[OFF_ADD];
            const float issub  = row[OFF_SUB];
            const bool proc = (active > 0.5f) && ((isadd > 0.5f) || (issub > 0.5f));
            const float ra = red[wave][half * 2 + 0][tok] + b2a;
            const float rs = red[wave][half * 2 + 1][tok] + b2s;
            const float r  = (isadd > 0.5f) ? ra : rs;
            float rr = rintf(r);                           // round-half-even == jnp.round
            rr = fminf(fmaxf(rr, 0.f), 15.f);
            const int res = (int)rr;
            if (proc) {
                const int obase = half ? OFF_OUTHI : OFF_OUTLO;
                myLds[tok * ROWP + obase + res] += 2.0f;   // one-hot * 2.0 scatter-add
            }
        }

        // ---- stream modified tokens back out (coalesced b128) ----
        #pragma unroll
        for (int i = 0; i < TOKS; ++i) {
            const float4 v = *(const float4*)(myLds + i * ROWP + lane * 4);
            *(float4*)(dst + i * D_BD + lane * 4) = v;
        }
    }
}

extern "C" void kernel_launch(void* const* d_in, const int* in_sizes, int n_in,
                              void* d_out, int out_size, void* d_ws, size_t ws_size,
                              hipStream_t stream) {
    (void)n_in; (void)d_ws; (void)ws_size; (void)out_size;
    const float* x   = (const float*)d_in[0];
    const float* Wa1 = (const float*)d_in[1];
    const float* ba1 = (const float*)d_in[2];
    const float* Wa2 = (const float*)d_in[3];
    const float* ba2 = (const float*)d_in[4];
    const float* Ws1 = (const float*)d_in[5];
    const float* bs1 = (const float*)d_in[6];
    const float* Ws2 = (const float*)d_in[7];
    const float* bs2 = (const float*)d_in[8];
    float* outp = (float*)d_out;

    const int numTokens = in_sizes[0] / D_BD;        // B*S = 65536
    const int numTiles  = numTokens / TOKS;          // 4096 wave-tiles

    const int blocks = 512;                          // 2048 waves, 2 tiles each
    alu_wmma_kernel<<<blocks, WAVES_PB * 32, 0, stream>>>(
        x, Wa1, ba1, Wa2, ba2, Ws1, bs1, Ws2, bs2, outp, numTiles);
}